// dHTTokenizer_23751169147439
// MI455X (gfx1250) — compile-verified
//
#include <hip/hip_runtime.h>

// Problem constants from the reference (H = W = 1024 -> plane = 1<<20).
#define HW_LOG2 20
#define HW_MASK ((1 << HW_LOG2) - 1)

#if __has_builtin(__builtin_amdgcn_global_load_async_to_lds_b128)
#define SEG_ASYNC 1
#else
#define SEG_ASYNC 0
#endif

typedef int vint4 __attribute__((vector_size(16)));            // native clang vector
typedef __attribute__((address_space(1))) void*  as1_voidp;
typedef __attribute__((address_space(3))) void*  as3_voidp;
typedef __attribute__((address_space(1))) vint4* as1_v4p;
typedef __attribute__((address_space(3))) vint4* as3_v4p;

__global__ __launch_bounds__(256) void zero_f32(float* __restrict__ p, int n) {
    int i = blockIdx.x * 256 + threadIdx.x;
    if (i < n) p[i] = 0.0f;
}

// Pass 1: scatter-reduce sums[NV][4] = {sum_c0, sum_c1, sum_c2, count}
__global__ __launch_bounds__(256) void accum_kernel(
    const float* __restrict__ img, const int* __restrict__ seg,
    float* __restrict__ sums, int nquad) {
    int gid = blockIdx.x * 256 + threadIdx.x;
    if (gid >= nquad) return;
    size_t i4 = (size_t)gid * 4;            // 4 consecutive pixels, same plane row
#if SEG_ASYNC
    __shared__ vint4 seg_tile[256];
    as1_v4p gsrc = (as1_v4p)(as1_voidp)(void*)(seg + i4);
    as3_v4p ldst = (as3_v4p)(as3_voidp)(void*)(&seg_tile[threadIdx.x]);
    __builtin_amdgcn_global_load_async_to_lds_b128(gsrc, ldst, 0, 0);
#endif
    int b  = (int)(i4 >> HW_LOG2);
    int sp = (int)(i4 & HW_MASK);
    const float* base = img + (((size_t)b * 3) << HW_LOG2) + sp;
    float4 p0 = *(const float4*)(base);
    float4 p1 = *(const float4*)(base + (1u << HW_LOG2));
    float4 p2 = *(const float4*)(base + (2u << HW_LOG2));
    int sx, sy, sz, sw;
#if SEG_ASYNC
#if __has_builtin(__builtin_amdgcn_s_wait_asynccnt)
    __builtin_amdgcn_s_wait_asynccnt(0);
#else
    asm volatile("s_wait_asynccnt 0" ::: "memory");
#endif
    vint4 sv = seg_tile[threadIdx.x];
    sx = sv[0]; sy = sv[1]; sz = sv[2]; sw = sv[3];
#else
    int4 s = *(const int4*)(seg + i4);
    sx = s.x; sy = s.y; sz = s.z; sw = s.w;
#endif
    float* a;
    a = sums + (size_t)sx * 4;
    atomicAdd(a + 0, p0.x); atomicAdd(a + 1, p1.x); atomicAdd(a + 2, p2.x); atomicAdd(a + 3, 1.0f);
    a = sums + (size_t)sy * 4;
    atomicAdd(a + 0, p0.y); atomicAdd(a + 1, p1.y); atomicAdd(a + 2, p2.y); atomicAdd(a + 3, 1.0f);
    a = sums + (size_t)sz * 4;
    atomicAdd(a + 0, p0.z); atomicAdd(a + 1, p1.z); atomicAdd(a + 2, p2.z); atomicAdd(a + 3, 1.0f);
    a = sums + (size_t)sw * 4;
    atomicAdd(a + 0, p0.w); atomicAdd(a + 1, p1.w); atomicAdd(a + 2, p2.w); atomicAdd(a + 3, 1.0f);
}

// Pass 2: rm4[v] = {fV[v] - sums[v]/max(cnt,1), pad}
__global__ __launch_bounds__(256) void means_kernel(
    const float* __restrict__ fV, const float* __restrict__ sums,
    float4* __restrict__ rm4, int nv) {
    int v = blockIdx.x * 256 + threadIdx.x;
    if (v >= nv) return;
    float c = fmaxf(sums[(size_t)v * 4 + 3], 1.0f);
    float4 o;
    o.x = fV[(size_t)v * 3 + 0] - sums[(size_t)v * 4 + 0] / c;
    o.y = fV[(size_t)v * 3 + 1] - sums[(size_t)v * 4 + 1] / c;
    o.z = fV[(size_t)v * 3 + 2] - sums[(size_t)v * 4 + 2] / c;
    o.w = 0.0f;
    rm4[v] = o;
}

// Pass 3: out[i,c] = pix[i,c] + rm[seg[i],c]   (3 aligned float4 stores / thread)
__global__ __launch_bounds__(256) void out_kernel(
    const float* __restrict__ img, const int* __restrict__ seg,
    const float4* __restrict__ rm4, float* __restrict__ out, int nquad) {
    int gid = blockIdx.x * 256 + threadIdx.x;
    if (gid >= nquad) return;
    size_t i4 = (size_t)gid * 4;
    int b  = (int)(i4 >> HW_LOG2);
    int sp = (int)(i4 & HW_MASK);
    const float* base = img + (((size_t)b * 3) << HW_LOG2) + sp;
    float4 p0 = *(const float4*)(base);
    float4 p1 = *(const float4*)(base + (1u << HW_LOG2));
    float4 p2 = *(const float4*)(base + (2u << HW_LOG2));
    int4 s = *(const int4*)(seg + i4);
    float4 r0 = rm4[s.x];
    float4 r1 = rm4[s.y];
    float4 r2 = rm4[s.z];
    float4 r3 = rm4[s.w];
    float4 v0 = make_float4(p0.x + r0.x, p1.x + r0.y, p2.x + r0.z, p0.y + r1.x);
    float4 v1 = make_float4(p1.y + r1.y, p2.y + r1.z, p0.z + r2.x, p1.z + r2.y);
    float4 v2 = make_float4(p2.z + r2.z, p0.w + r3.x, p1.w + r3.y, p2.w + r3.z);
    float4* o = (float4*)(out + (size_t)gid * 12);   // 48*gid bytes, 16B aligned
    o[0] = v0; o[1] = v1; o[2] = v2;
}

extern "C" void kernel_launch(void* const* d_in, const int* in_sizes, int n_in,
                              void* d_out, int out_size, void* d_ws, size_t ws_size,
                              hipStream_t stream) {
    const float* img = (const float*)d_in[0];
    const int*   seg = (const int*)d_in[1];
    const float* fV  = (const float*)d_in[2];
    // derive sizes: N = #pixels, C = channels, NV = #segments
    const int N  = in_sizes[1];
    const int C  = in_sizes[0] / N;        // = 3
    const int NV = in_sizes[2] / C;        // = 29000

    float*  sums = (float*)d_ws;                          // NV*4 floats
    float4* rm4  = (float4*)(sums + (size_t)NV * 4);      // NV float4 (16B aligned)

    const int nzero = NV * 4;
    zero_f32<<<(nzero + 255) / 256, 256, 0, stream>>>(sums, nzero);

    const int nquad = N / 4;               // N = 16 * 2^20, divisible by 4
    accum_kernel<<<(nquad + 255) / 256, 256, 0, stream>>>(img, seg, sums, nquad);
    means_kernel<<<(NV + 255) / 256, 256, 0, stream>>>(fV, sums, rm4, NV);
    out_kernel<<<(nquad + 255) / 256, 256, 0, stream>>>(img, seg, rm4, (float*)d_out, nquad);
}